// adaptive_aggregation_23313082482951
// MI455X (gfx1250) — compile-verified
//
#include <hip/hip_runtime.h>
#include <hip/hip_bf16.h>

// Adaptive aggregation: out[b,h,w] = sum_{7x7} max(thick,thin)[h+dh-3,w+dw-3] * coeff[b,j,h,w]
// Bandwidth-bound (agg_coeff = 411 MB streamed once; intensity ~0.5 FLOP/B -> ~19us floor @ 23.3 TB/s).
// CDNA5 paths: async global->LDS halo staging (ASYNCcnt), global_prefetch_b8 on the
// dominant stream, NT b128 coeff loads / NT b128 stores.

#define KK      7
#define PAD     3
#define K2      49
#define TILE    32
#define HALO    38                 // TILE + KK - 1
#define LDSTR   39                 // odd stride -> the 4 ty-classes of a wave hit disjoint
                                   // bank classes mod 4 -> conflict-free ds_load
#define IMG_H   512
#define IMG_W   512
#define IMG_B   8
#define HW      (IMG_H * IMG_W)

typedef float vfloat4 __attribute__((ext_vector_type(4)));

__global__ __launch_bounds__(256) void adaptive_agg_kernel(
    const float* __restrict__ thick,
    const float* __restrict__ thin,
    const float* __restrict__ coeff,
    float* __restrict__ out)
{
    __shared__ float lds_a[HALO * LDSTR];   // thick halo -> becomes sal after max-pass
    __shared__ float lds_b[HALO * LDSTR];   // thin halo

    const int tx  = threadIdx.x;            // 0..7  (x quads)
    const int ty  = threadIdx.y;            // 0..31 (rows)
    const int tid = ty * 8 + tx;            // 0..255
    const int x0t = blockIdx.x * TILE;
    const int y0t = blockIdx.y * TILE;
    const int b   = blockIdx.z;

    const float* thick_b = thick + (size_t)b * HW;
    const float* thin_b  = thin  + (size_t)b * HW;

    // Per-thread coeff stream base (computed early so we can prefetch it).
    const int lx0 = tx * 4;                 // 0..28, 16B-aligned quads
    const int ly  = ty;
    const int x   = x0t + lx0;
    const int y   = y0t + ly;
    const float* cp = coeff + ((size_t)b * K2) * HW + (size_t)y * IMG_W + x;

    // ---- Warm the first coeff planes (global_prefetch_b8) while the halo DMA runs ----
#pragma unroll
    for (int j = 0; j < KK; ++j)
        __builtin_prefetch(cp + (size_t)j * HW, 0, 0);

    // ---- Stage 38x38 halos of thick & thin into LDS via async global->LDS DMA ----
    // Per-lane b32 copies; addresses clamped to the image (padding fixed to 0 below).
    for (int i = tid; i < HALO * HALO; i += 256) {
        const int r  = i / HALO;
        const int c  = i - r * HALO;
        const int gy = y0t - PAD + r;
        const int gx = x0t - PAD + c;
        const int cy = min(max(gy, 0), IMG_H - 1);
        const int cx = min(max(gx, 0), IMG_W - 1);
        const unsigned voff    = (unsigned)(cy * IMG_W + cx) * 4u;
        const unsigned lds_off = (unsigned)(r * LDSTR + c) * 4u;
        // low 32 bits of a flat LDS address == byte offset the DS/async HW expects
        const unsigned la = (unsigned)(uintptr_t)(&lds_a[0]) + lds_off;
        const unsigned lb = (unsigned)(uintptr_t)(&lds_b[0]) + lds_off;
        asm volatile("global_load_async_to_lds_b32 %0, %1, %2"
                     :: "v"(la), "v"(voff), "s"(thick_b) : "memory");
        asm volatile("global_load_async_to_lds_b32 %0, %1, %2"
                     :: "v"(lb), "v"(voff), "s"(thin_b) : "memory");
    }
    asm volatile("s_wait_asynccnt 0x0" ::: "memory");  // wave-local async completion
    __syncthreads();                                   // workgroup-wide visibility

    // ---- Fuse sal = max(thick, thin) in LDS; zero the out-of-image padding ----
    for (int i = tid; i < HALO * HALO; i += 256) {
        const int r  = i / HALO;
        const int c  = i - r * HALO;
        const int gy = y0t - PAD + r;
        const int gx = x0t - PAD + c;
        const bool valid = (gy >= 0) & (gy < IMG_H) & (gx >= 0) & (gx < IMG_W);
        const int li = r * LDSTR + c;
        const float v = fmaxf(lds_a[li], lds_b[li]);
        lds_a[li] = valid ? v : 0.0f;
    }
    __syncthreads();

    // ---- Main loop: stream coeff (NT b128), FMA against LDS sal ----
    float acc0 = 0.f, acc1 = 0.f, acc2 = 0.f, acc3 = 0.f;

#pragma unroll
    for (int dh = 0; dh < KK; ++dh) {
        // 10 sal values cover dw=0..6 for 4 consecutive pixels
        float s[10];
        const float* srow = &lds_a[(ly + dh) * LDSTR + lx0];
#pragma unroll
        for (int m = 0; m < 10; ++m) s[m] = srow[m];

#pragma unroll
        for (int dw = 0; dw < KK; ++dw) {
            const int j = dh * KK + dw;
            const vfloat4 c4 = __builtin_nontemporal_load(
                reinterpret_cast<const vfloat4*>(cp + (size_t)j * HW));
            acc0 = fmaf(c4.x, s[dw + 0], acc0);
            acc1 = fmaf(c4.y, s[dw + 1], acc1);
            acc2 = fmaf(c4.z, s[dw + 2], acc2);
            acc3 = fmaf(c4.w, s[dw + 3], acc3);
        }
    }

    vfloat4 o = { acc0, acc1, acc2, acc3 };
    __builtin_nontemporal_store(o,
        reinterpret_cast<vfloat4*>(out + (size_t)b * HW + (size_t)y * IMG_W + x));
}

extern "C" void kernel_launch(void* const* d_in, const int* in_sizes, int n_in,
                              void* d_out, int out_size, void* d_ws, size_t ws_size,
                              hipStream_t stream) {
    const float* thick = (const float*)d_in[0];   // [8,1,512,512] f32
    const float* thin  = (const float*)d_in[1];   // [8,1,512,512] f32
    const float* coeff = (const float*)d_in[2];   // [8,49,512,512] f32
    float* out = (float*)d_out;                   // [8,1,512,512] f32

    dim3 block(8, 32, 1);                         // 256 threads = 8 waves (wave32)
    dim3 grid(IMG_W / TILE, IMG_H / TILE, IMG_B); // 16 x 16 x 8 = 2048 tiles
    adaptive_agg_kernel<<<grid, block, 0, stream>>>(thick, thin, coeff, out);
}